// Pers2Equi_59785944760704
// MI455X (gfx1250) — compile-verified
//
#include <hip/hip_runtime.h>
#include <stdint.h>

#define ERP_PIX   (256 * 512)   // 131072 output pixels
#define NPATCH    18
#define CCH       32            // channels == wave32 lanes
#define PH        224
#define PW        224
#define CH_STRIDE (PH * PW * NPATCH)   // 903168 elements between channels
#define PIX_PER_BLOCK 32
#define THREADS   256

__global__ __launch_bounds__(THREADS)
void pers2equi_kernel(const float* __restrict__ img,     // (1,32,224,224,18)
                      const float* __restrict__ wlist,   // (18,256,512,4)
                      const float* __restrict__ maskp,   // (18,256,512)
                      const int*   __restrict__ y0p,
                      const int*   __restrict__ x0p,
                      const int*   __restrict__ y1p,
                      const int*   __restrict__ x1p,
                      float*       __restrict__ out)     // (1,32,256,512)
{
    __shared__ __align__(16) float s_w[PIX_PER_BLOCK][NPATCH][4];
    __shared__             float s_m[PIX_PER_BLOCK][NPATCH];
    __shared__ __align__(16) int   s_i[PIX_PER_BLOCK][NPATCH][4]; // y0,x0,y1,x1
    __shared__             float s_out[CCH][PIX_PER_BLOCK + 1];   // +1: bank-conflict pad

    const int tid     = threadIdx.x;
    const int pixbase = blockIdx.x * PIX_PER_BLOCK;

    // ------- Stage metadata into LDS with CDNA5 async global->LDS DMA -------
    // Streams are read exactly once -> TH_LOAD_NT so they don't evict the
    // L2-resident pers_img (116 MB of 192 MB L2) that the gathers reuse.
    for (int v = tid; v < PIX_PER_BLOCK * NPATCH; v += THREADS) {
        const int p  = v >> 5;          // PIX_PER_BLOCK == 32
        const int px = v & 31;
        const uint32_t o = (uint32_t)p * ERP_PIX + (uint32_t)(pixbase + px);

        uint32_t d_w = (uint32_t)(size_t)&s_w[px][p][0];
        unsigned long long g_w = (unsigned long long)(size_t)(wlist + 4u * (size_t)o);
        asm volatile("global_load_async_to_lds_b128 %0, %1, off th:TH_LOAD_NT"
                     :: "v"(d_w), "v"(g_w) : "memory");

        uint32_t d_m = (uint32_t)(size_t)&s_m[px][p];
        unsigned long long g_m = (unsigned long long)(size_t)(maskp + o);
        asm volatile("global_load_async_to_lds_b32 %0, %1, off th:TH_LOAD_NT"
                     :: "v"(d_m), "v"(g_m) : "memory");

        uint32_t d_i = (uint32_t)(size_t)&s_i[px][p][0];
        unsigned long long g0 = (unsigned long long)(size_t)(y0p + o);
        unsigned long long g1 = (unsigned long long)(size_t)(x0p + o);
        unsigned long long g2 = (unsigned long long)(size_t)(y1p + o);
        unsigned long long g3 = (unsigned long long)(size_t)(x1p + o);
        asm volatile("global_load_async_to_lds_b32 %0, %1, off th:TH_LOAD_NT"
                     :: "v"(d_i), "v"(g0) : "memory");
        asm volatile("global_load_async_to_lds_b32 %0, %1, off th:TH_LOAD_NT"
                     :: "v"(d_i + 4u), "v"(g1) : "memory");
        asm volatile("global_load_async_to_lds_b32 %0, %1, off th:TH_LOAD_NT"
                     :: "v"(d_i + 8u), "v"(g2) : "memory");
        asm volatile("global_load_async_to_lds_b32 %0, %1, off th:TH_LOAD_NT"
                     :: "v"(d_i + 12u), "v"(g3) : "memory");
    }
    asm volatile("s_wait_asynccnt 0" ::: "memory");
    __syncthreads();

    // ------- Compute: lane = channel (wave32), each wave owns 4 pixels -------
    const int lane = tid & 31;   // channel
    const int wave = tid >> 5;   // 0..7
    const float* __restrict__ imgc = img + (size_t)lane * CH_STRIDE;

    for (int i = 0; i < 4; ++i) {
        const int px = wave * 4 + i;

        // per-pixel denominator: sum over (patch, corner) of thresholded weights
        float den = 0.0f;
#pragma unroll
        for (int p = 0; p < NPATCH; ++p) {
            const float4 wv = *(const float4*)&s_w[px][p][0];
            den += (wv.x > 1e-5f ? wv.x : 0.0f)
                 + (wv.y > 1e-5f ? wv.y : 0.0f)
                 + (wv.z > 1e-5f ? wv.z : 0.0f)
                 + (wv.w > 1e-5f ? wv.w : 0.0f);
        }
        const float rden = 1.0f / fmaxf(den, 1e-12f);

        float acc = 0.0f;
#pragma unroll
        for (int p = 0; p < NPATCH; ++p) {
            const float4 wv = *(const float4*)&s_w[px][p][0];
            const float  m  = s_m[px][p] * rden;
            const float wa = (wv.x > 1e-5f ? wv.x : 0.0f) * m;
            const float wb = (wv.y > 1e-5f ? wv.y : 0.0f) * m;
            const float wc = (wv.z > 1e-5f ? wv.z : 0.0f) * m;
            const float wd = (wv.w > 1e-5f ? wv.w : 0.0f) * m;
            // wave-uniform skip of masked-out patches (s_cbranch_execz)
            if ((wa != 0.0f) | (wb != 0.0f) | (wc != 0.0f) | (wd != 0.0f)) {
                const int4 ix = *(const int4*)&s_i[px][p][0]; // y0,x0,y1,x1
                const uint32_t r0 = (uint32_t)ix.x * (PW * NPATCH);
                const uint32_t r1 = (uint32_t)ix.z * (PW * NPATCH);
                const uint32_t c0 = (uint32_t)ix.y * NPATCH + (uint32_t)p;
                const uint32_t c1 = (uint32_t)ix.w * NPATCH + (uint32_t)p;
                acc = fmaf(wa, imgc[r0 + c0], acc);  // Ia: (y0,x0)
                acc = fmaf(wb, imgc[r1 + c0], acc);  // Ib: (y1,x0)
                acc = fmaf(wc, imgc[r0 + c1], acc);  // Ic: (y0,x1)
                acc = fmaf(wd, imgc[r1 + c1], acc);  // Id: (y1,x1)
            }
        }
        s_out[lane][px] = acc;
    }
    __syncthreads();

    // ------- Coalesced, non-temporal writeback via LDS transpose -------
    for (int i = 0; i < 4; ++i) {
        const int v  = tid + i * THREADS;   // 0..1023
        const int c  = v >> 5;              // channel
        const int px = v & 31;              // pixel within tile
        __builtin_nontemporal_store(s_out[c][px],
                                    out + (size_t)c * ERP_PIX + (size_t)(pixbase + px));
    }
}

extern "C" void kernel_launch(void* const* d_in, const int* in_sizes, int n_in,
                              void* d_out, int out_size, void* d_ws, size_t ws_size,
                              hipStream_t stream) {
    (void)in_sizes; (void)n_in; (void)out_size; (void)d_ws; (void)ws_size;
    // setup_inputs order: pers_img, w_list, mask, y0, x0, y1, x1
    const float* img   = (const float*)d_in[0];
    const float* wlist = (const float*)d_in[1];
    const float* maskp = (const float*)d_in[2];
    const int*   y0p   = (const int*)d_in[3];
    const int*   x0p   = (const int*)d_in[4];
    const int*   y1p   = (const int*)d_in[5];
    const int*   x1p   = (const int*)d_in[6];
    float* out = (float*)d_out;

    const int blocks = ERP_PIX / PIX_PER_BLOCK; // 4096
    pers2equi_kernel<<<blocks, THREADS, 0, stream>>>(img, wlist, maskp,
                                                     y0p, x0p, y1p, x1p, out);
}